// YOLOV8LabelEncoder_44994077393560
// MI455X (gfx1250) — compile-verified
//
#include <hip/hip_runtime.h>
#include <math.h>
#include <stdint.h>

#define TPB 256
#define KSEL 10

// ---- optional CDNA5 async global->LDS path (compile-safe guards) ----
#if defined(__HIP_DEVICE_COMPILE__) && defined(__gfx1250__) && \
    __has_builtin(__builtin_amdgcn_global_load_async_to_lds_b32) && \
    __has_builtin(__builtin_amdgcn_s_wait_asynccnt)
#define USE_ASYNC_LDS 1
typedef __attribute__((address_space(1))) int as1_int;
typedef __attribute__((address_space(3))) int as3_int;
#endif

__device__ __forceinline__ float pow6f(float x) { float x2 = x * x; return x2 * x2 * x2; }

// Matches reference compute_ciou followed by max(.,0). atan terms precomputed.
__device__ __forceinline__ float ciou_cl(
    float x1a, float y1a, float x2a, float y2a,   // gt
    float x1b, float y1b, float x2b, float y2b,   // pred
    float atan_gt, float atan_pd)
{
    const float E = 1e-7f;
    float w1 = x2a - x1a, h1 = y2a - y1a + E;
    float w2 = x2b - x1b, h2 = y2b - y1b + E;
    float iw = fmaxf(fminf(x2a, x2b) - fmaxf(x1a, x1b), 0.0f);
    float ih = fmaxf(fminf(y2a, y2b) - fmaxf(y1a, y1b), 0.0f);
    float inter = iw * ih;
    float uni = w1 * h1 + w2 * h2 - inter;
    float iou = inter / (uni + E);
    float cw = fmaxf(x2a, x2b) - fminf(x1a, x1b);
    float ch = fmaxf(y2a, y2b) - fminf(y1a, y1b);
    float c2 = cw * cw + ch * ch + E;
    float dx = x1b + x2b - x1a - x2a;
    float dy = y1b + y2b - y1a - y2a;
    float rho2 = (dx * dx + dy * dy) * 0.25f;
    float da = atan_pd - atan_gt;
    float v = (4.0f / (float)(M_PI * M_PI)) * da * da;
    float al = v / (v - iou + (1.0f + E));
    return fmaxf(iou - (rho2 / c2 + v * al), 0.0f);
}

// K0: precompute atan(w2/h2) per pred box; zero per-call scratch.
__global__ void k0_init(const float* __restrict__ pdb, float* __restrict__ atanb,
                        int* __restrict__ cnt, int* __restrict__ msum,
                        unsigned int* __restrict__ posAl, unsigned int* __restrict__ posOv,
                        int BA, int BM)
{
    int i = blockIdx.x * blockDim.x + threadIdx.x;
    int stride = gridDim.x * blockDim.x;
    for (; i < BA; i += stride) {
        float4 pb = ((const float4*)pdb)[i];
        float w2 = pb.z - pb.x;
        float h2 = pb.w - pb.y + 1e-7f;
        atanb[i] = atanf(w2 / h2);
        cnt[i] = 0;
        msum[i] = 0;
        if (i < BM) { posAl[i] = 0u; posOv[i] = 0u; }
    }
}

// K2: per (b,m) block — align_metric over A into LDS, iterative top-10,
// count valid selections into cnt/msum.
__global__ void k2_topk(const float* __restrict__ pds, const float* __restrict__ pdb,
                        const float* __restrict__ anc, const float* __restrict__ gl,
                        const float* __restrict__ gtb, const unsigned char* __restrict__ mgt,
                        const float* __restrict__ atanb,
                        int* __restrict__ cnt, int* __restrict__ msum,
                        int A, int C, int M)
{
    extern __shared__ float smem[];
    int bm = blockIdx.x;
    if (!mgt[bm]) return;                // masked gt row contributes nothing
    int b = bm / M, m = bm % M;
    float* smet = smem;                  // A floats
    float* redv = smem + A;              // TPB floats
    int*   redi = (int*)(redv + TPB);    // TPB ints
    int*   sel  = redi + TPB;            // KSEL ints
    int tid = threadIdx.x;

    const float4 g = ((const float4*)gtb)[bm];
    float x1a = g.x, y1a = g.y, x2a = g.z, y2a = g.w;
    float w1 = x2a - x1a, h1 = y2a - y1a + 1e-7f;
    float atan_gt = atanf(w1 / h1);
    int lbl = (int)gl[bm];
    size_t baseA = (size_t)b * A;
    const float4* pbB = (const float4*)pdb + baseA;
    const float*  psB = pds + baseA * (size_t)C;
    const float*  atB = atanb + baseA;
    const float2* anc2 = (const float2*)anc;

    for (int a = tid; a < A; a += TPB) {
        if (a + TPB < A) {
            __builtin_prefetch(&pbB[a + TPB], 0, 0);            // global_prefetch_b8
            if (lbl >= 0) __builtin_prefetch(&psB[(size_t)(a + TPB) * C + lbl], 0, 0);
        }
        float2 ap = anc2[a];
        float d0 = ap.x - x1a, d1 = ap.y - y1a, d2 = x2a - ap.x, d3 = y2a - ap.y;
        bool valid = fminf(fminf(d0, d1), fminf(d2, d3)) > 1e-9f;
        float met = 0.0f;
        if (valid) {
            float4 pb = pbB[a];
            float sc = (lbl >= 0) ? psB[(size_t)a * C + lbl] : 0.0f;
            float ov = ciou_cl(x1a, y1a, x2a, y2a, pb.x, pb.y, pb.z, pb.w, atan_gt, atB[a]);
            met = sqrtf(sc) * pow6f(ov);
        }
        smet[a] = met;
    }
    __syncthreads();

    // 10 rounds of stable block argmax (lowest index on ties == XLA top_k)
    for (int r = 0; r < KSEL; ++r) {
        float bv = -1.0f; int bi = 0x7fffffff;
        for (int a = tid; a < A; a += TPB) {
            float v = smet[a];
            if (v > bv) { bv = v; bi = a; }
        }
        redv[tid] = bv; redi[tid] = bi;
        __syncthreads();
        for (int s = TPB / 2; s > 0; s >>= 1) {
            if (tid < s) {
                float vo = redv[tid + s]; int io = redi[tid + s];
                if (vo > redv[tid] || (vo == redv[tid] && io < redi[tid])) {
                    redv[tid] = vo; redi[tid] = io;
                }
            }
            __syncthreads();
        }
        if (tid == 0) { sel[r] = redi[0]; smet[redi[0]] = -1.0f; }
        __syncthreads();
    }

    if (tid < KSEL) {
        int a = sel[tid];
        float2 ap = anc2[a];
        float d0 = ap.x - x1a, d1 = ap.y - y1a, d2 = x2a - ap.x, d3 = y2a - ap.y;
        if (fminf(fminf(d0, d1), fminf(d2, d3)) > 1e-9f) {   // mask_in_gts
            atomicAdd(&cnt[baseA + a], 1);
            atomicAdd(&msum[baseA + a], m);
        }
    }
}

// K3: per-anchor resolution of mask_pos (incl. multi-gt argmax-overlap fixup),
// emits target_bboxes, fg_mask, align@tgi; atomic-max of pos_align/pos_overlaps.
__global__ void k3_resolve(const float* __restrict__ pds, const float* __restrict__ pdb,
                           const float* __restrict__ anc, const float* __restrict__ gl,
                           const float* __restrict__ gtb, const unsigned char* __restrict__ mgt,
                           const float* __restrict__ atanb,
                           const int* __restrict__ cnt, const int* __restrict__ msum,
                           int* __restrict__ tgiArr, float* __restrict__ amArr,
                           unsigned int* __restrict__ posAl, unsigned int* __restrict__ posOv,
                           float* __restrict__ outBB, float* __restrict__ outFG,
                           int A, int C, int M)
{
    __shared__ float sgt[64 * 4];
    __shared__ float satan[64];
    __shared__ unsigned char smk[64];
    int b = blockIdx.y;
    int tid = threadIdx.x;
    int M4 = M * 4;

    if (tid < M4) {
#if defined(USE_ASYNC_LDS)
        __builtin_amdgcn_global_load_async_to_lds_b32(
            (as1_int*)(uintptr_t)(gtb + (size_t)b * M4 + tid),
            (as3_int*)(&sgt[tid]), 0, 0);
#else
        sgt[tid] = gtb[(size_t)b * M4 + tid];
#endif
    }
#if defined(USE_ASYNC_LDS)
    __builtin_amdgcn_s_wait_asynccnt(0);
#endif
    __syncthreads();
    if (tid < M) {
        float w1 = sgt[tid * 4 + 2] - sgt[tid * 4 + 0];
        float h1 = sgt[tid * 4 + 3] - sgt[tid * 4 + 1] + 1e-7f;
        satan[tid] = atanf(w1 / h1);
        smk[tid] = mgt[(size_t)b * M + tid];
    }
    __syncthreads();

    int a = blockIdx.x * TPB + tid;
    if (a >= A) return;
    size_t ba = (size_t)b * A + a;
    int c = cnt[ba];
    int tg = 0; float am = 0.0f, ov = 0.0f; int fg = 0;
    if (c > 0) {
        fg = 1;
        float4 pb = ((const float4*)pdb)[ba];
        float at2 = atanb[ba];
        if (c == 1) {
            tg = msum[ba];
            ov = ciou_cl(sgt[tg*4], sgt[tg*4+1], sgt[tg*4+2], sgt[tg*4+3],
                         pb.x, pb.y, pb.z, pb.w, satan[tg], at2);
        } else {
            // multi-assigned: is_max = one_hot(argmax_m overlaps) (first max)
            float2 ap = ((const float2*)anc)[a];
            float best = -1.0f; int bi = 0;
            for (int mm = 0; mm < M; ++mm) {
                float x1 = sgt[mm*4], y1 = sgt[mm*4+1], x2 = sgt[mm*4+2], y2 = sgt[mm*4+3];
                float o = 0.0f;
                float d0 = ap.x - x1, d1 = ap.y - y1, d2 = x2 - ap.x, d3 = y2 - ap.y;
                bool valid = (fminf(fminf(d0, d1), fminf(d2, d3)) > 1e-9f) && smk[mm];
                if (valid)
                    o = ciou_cl(x1, y1, x2, y2, pb.x, pb.y, pb.z, pb.w, satan[mm], at2);
                if (o > best) { best = o; bi = mm; }
            }
            tg = bi; ov = fmaxf(best, 0.0f);
        }
        int lbl = (int)gl[(size_t)b * M + tg];
        float sc = (lbl >= 0) ? pds[ba * (size_t)C + lbl] : 0.0f;
        am = sqrtf(sc) * pow6f(ov);   // ov==0 whenever (tg,a) invalid -> am==0, matches ref
        atomicMax(&posAl[(size_t)b * M + tg], __float_as_uint(am));
        atomicMax(&posOv[(size_t)b * M + tg], __float_as_uint(ov));
    }
    tgiArr[ba] = tg;
    amArr[ba] = am;
    ((float4*)outBB)[ba] = make_float4(sgt[tg*4], sgt[tg*4+1], sgt[tg*4+2], sgt[tg*4+3]);
    outFG[ba] = fg ? 1.0f : 0.0f;
}

// K4: target_scores = one_hot(label) * norm, fg-gated; fully coalesced writes.
__global__ void k4_scores(const float* __restrict__ gl,
                          const int* __restrict__ cnt, const int* __restrict__ tgiArr,
                          const float* __restrict__ amArr,
                          const unsigned int* __restrict__ posAl, const unsigned int* __restrict__ posOv,
                          float* __restrict__ outSc,
                          int A, int C, int M, long long total)
{
    long long i = (long long)blockIdx.x * blockDim.x + threadIdx.x;
    long long stride = (long long)gridDim.x * blockDim.x;
    for (; i < total; i += stride) {
        int cch = (int)(i % C);
        long long ba = i / C;
        int b = (int)(ba / A);
        float out = 0.0f;
        if (cnt[ba] > 0) {
            int tg = tgiArr[ba];
            size_t bm = (size_t)b * M + tg;
            int lbl = (int)fmaxf(gl[bm], 0.0f);
            if (cch == lbl) {
                float pa = __uint_as_float(posAl[bm]);
                float po = __uint_as_float(posOv[bm]);
                out = amArr[ba] * po / (pa + 1e-9f);
            }
        }
        outSc[i] = out;
    }
}

extern "C" void kernel_launch(void* const* d_in, const int* in_sizes, int n_in,
                              void* d_out, int out_size, void* d_ws, size_t ws_size,
                              hipStream_t stream)
{
    const float* pds = (const float*)d_in[0];            // (B,A,C)
    const float* pdb = (const float*)d_in[1];            // (B,A,4)
    const float* anc = (const float*)d_in[2];            // (A,2)
    const float* gl  = (const float*)d_in[3];            // (B,M)
    const float* gtb = (const float*)d_in[4];            // (B,M,4)
    const unsigned char* mgt = (const unsigned char*)d_in[5]; // (B,M,1) bool

    int A = in_sizes[2] / 2;
    int B = in_sizes[1] / (A * 4);
    int C = (int)(in_sizes[0] / ((long long)B * A));
    int M = in_sizes[3] / B;
    int BA = B * A, BM = B * M;

    int* cnt   = (int*)d_ws;                      // BA
    int* msum  = cnt + BA;                        // BA
    int* tgi   = msum + BA;                       // BA
    float* am  = (float*)(tgi + BA);              // BA
    float* atanb = am + BA;                       // BA
    unsigned int* posAl = (unsigned int*)(atanb + BA); // BM
    unsigned int* posOv = posAl + BM;                  // BM

    float* outBB = (float*)d_out;
    float* outSc = outBB + (size_t)BA * 4;
    float* outFG = outSc + (size_t)BA * C;

    k0_init<<<(BA + TPB - 1) / TPB, TPB, 0, stream>>>(pdb, atanb, cnt, msum, posAl, posOv, BA, BM);

    size_t shm = (size_t)A * sizeof(float) + TPB * sizeof(float) + TPB * sizeof(int) + KSEL * sizeof(int);
    k2_topk<<<BM, TPB, shm, stream>>>(pds, pdb, anc, gl, gtb, mgt, atanb, cnt, msum, A, C, M);

    dim3 g3((A + TPB - 1) / TPB, B);
    k3_resolve<<<g3, TPB, 0, stream>>>(pds, pdb, anc, gl, gtb, mgt, atanb, cnt, msum,
                                       tgi, am, posAl, posOv, outBB, outFG, A, C, M);

    long long total = (long long)BA * C;
    k4_scores<<<(int)((total + TPB - 1) / TPB), TPB, 0, stream>>>(gl, cnt, tgi, am, posAl, posOv,
                                                                  outSc, A, C, M, total);
}